// GAT_26121991094296
// MI455X (gfx1250) — compile-verified
//
#include <hip/hip_runtime.h>
#include <math.h>

#define IN_DIM   256
#define HEADS    3
#define HID      3
#define NCLS     3
#define NEG_SLOPE 0.2f
#define GAT_EPS  1e-16f

typedef __attribute__((ext_vector_type(2))) float v2f;
typedef __attribute__((ext_vector_type(8))) float v8f;

// ---- order-preserving float <-> uint encoding for atomicMax on floats ----
__device__ __forceinline__ unsigned fenc(float f) {
    unsigned u = __float_as_uint(f);
    return (u & 0x80000000u) ? ~u : (u | 0x80000000u);
}
__device__ __forceinline__ float fdec(unsigned u) {
    return (u & 0x80000000u) ? __uint_as_float(u & 0x7fffffffu)
                             : __uint_as_float(~u);
}
__device__ __forceinline__ float lrelu(float x) { return x > 0.f ? x : NEG_SLOPE * x; }

// =====================================================================
// Kernel 1: h = x @ W1  via V_WMMA_F32_16X16X4_F32.
// x: [N,256] row-major, W: [256,9] row-major, h: [N,9].
// One wave per 16-row tile; K loop of 64 WMMA steps; W padded to 16 cols in LDS.
// =====================================================================
__global__ __launch_bounds__(256) void gemm1_wmma(const float* __restrict__ x,
                                                  const float* __restrict__ W,
                                                  float* __restrict__ h, int N) {
    __shared__ float Wp[IN_DIM * 16];           // 16 KB, zero-padded cols 9..15
    for (int i = threadIdx.x; i < IN_DIM * 16; i += 256) {
        int k = i >> 4, n = i & 15;
        Wp[i] = (n < HEADS * HID) ? W[k * (HEADS * HID) + n] : 0.0f;
    }
    __syncthreads();

    int lane  = threadIdx.x & 31;
    int wave  = threadIdx.x >> 5;
    long tile = (long)blockIdx.x * 8 + wave;
    long m0   = tile * 16;
    if (m0 >= N) return;                        // wave-uniform: EXEC stays all-ones

    int laneM = lane & 15;
    int koff  = (lane < 16) ? 0 : 2;            // A/B VGPR layout: K and K+2 split
    const float* xrow = x + (m0 + laneM) * IN_DIM;

    v8f c = {};
    #pragma unroll 4
    for (int k0 = 0; k0 < IN_DIM; k0 += 4) {
        v2f a, b;
        a.x = xrow[k0 + koff];                  // (M=laneM, K=k0+koff)
        a.y = xrow[k0 + koff + 1];              // (M=laneM, K=k0+koff+1)
        b.x = Wp[(k0 + koff)     * 16 + laneM]; // (K=k0+koff,   N=laneM)
        b.y = Wp[(k0 + koff + 1) * 16 + laneM]; // (K=k0+koff+1, N=laneM)
        c = __builtin_amdgcn_wmma_f32_16x16x4_f32(false, a, false, b,
                                                  (short)0, c, false, false);
    }

    // C/D layout: VGPR v holds (M = v + (lane<16?0:8), N = lane%16)
    int rbase = (lane < 16) ? 0 : 8;
    if (laneM < HEADS * HID) {
        #pragma unroll
        for (int v = 0; v < 8; ++v)
            h[(m0 + rbase + v) * (HEADS * HID) + laneM] = c[v];
    }
}

// =====================================================================
// Kernel 2 (both layers): per-node attention dots + self-loop max init.
// h: [N,9], atts/attd: [3,3]; writes as_,ad_: [N,3], m: [N,3] (encoded).
// =====================================================================
__global__ void node_att(const float* __restrict__ h,
                         const float* __restrict__ atts,
                         const float* __restrict__ attd,
                         float* __restrict__ as_, float* __restrict__ ad_,
                         unsigned* __restrict__ m, int N) {
    int n = blockIdx.x * blockDim.x + threadIdx.x;
    if (n >= N) return;
    #pragma unroll
    for (int hh = 0; hh < HEADS; ++hh) {
        float s = 0.f, d = 0.f;
        #pragma unroll
        for (int cc = 0; cc < 3; ++cc) {
            float v = h[n * 9 + hh * 3 + cc];
            s += v * atts[hh * 3 + cc];
            d += v * attd[hh * 3 + cc];
        }
        as_[n * 3 + hh] = s;
        ad_[n * 3 + hh] = d;
        m[n * 3 + hh] = fenc(lrelu(s + d));     // self-loop edge seeds the max
    }
}

// =====================================================================
// Kernel 3: edge pass 1 — atomic max of attention logits per dst.
// =====================================================================
__global__ void edge_max(const int* __restrict__ src, const int* __restrict__ dst,
                         const float* __restrict__ as_, const float* __restrict__ ad_,
                         unsigned* __restrict__ m, int E) {
    int i = blockIdx.x * blockDim.x + threadIdx.x;
    if (i >= E) return;
    int s = src[i], d = dst[i];
    #pragma unroll
    for (int hh = 0; hh < HEADS; ++hh) {
        float e = lrelu(as_[s * 3 + hh] + ad_[d * 3 + hh]);
        atomicMax(&m[d * 3 + hh], fenc(e));
    }
}

// =====================================================================
// Kernel 4: node init of s/acc with the self-loop contribution.
// =====================================================================
__global__ void node_init(const float* __restrict__ h,
                          const float* __restrict__ as_, const float* __restrict__ ad_,
                          const unsigned* __restrict__ m,
                          float* __restrict__ s, float* __restrict__ acc, int N) {
    int n = blockIdx.x * blockDim.x + threadIdx.x;
    if (n >= N) return;
    #pragma unroll
    for (int hh = 0; hh < HEADS; ++hh) {
        float e  = lrelu(as_[n * 3 + hh] + ad_[n * 3 + hh]);
        float p  = expf(e - fdec(m[n * 3 + hh]));
        s[n * 3 + hh] = p;
        #pragma unroll
        for (int cc = 0; cc < 3; ++cc)
            acc[n * 9 + hh * 3 + cc] = p * h[n * 9 + hh * 3 + cc];
    }
}

// =====================================================================
// Kernel 5: edge pass 2 — fused exp-sum and unnormalized weighted scatter.
// =====================================================================
__global__ void edge_scatter(const int* __restrict__ src, const int* __restrict__ dst,
                             const float* __restrict__ h,
                             const float* __restrict__ as_, const float* __restrict__ ad_,
                             const unsigned* __restrict__ m,
                             float* __restrict__ s, float* __restrict__ acc, int E) {
    int i = blockIdx.x * blockDim.x + threadIdx.x;
    if (i >= E) return;
    int sn = src[i], dn = dst[i];
    __builtin_prefetch(&h[sn * 9], 0, 0);       // global_prefetch_b8 on gathered row
    #pragma unroll
    for (int hh = 0; hh < HEADS; ++hh) {
        float e = lrelu(as_[sn * 3 + hh] + ad_[dn * 3 + hh]);
        float p = expf(e - fdec(m[dn * 3 + hh]));
        atomicAdd(&s[dn * 3 + hh], p);
        #pragma unroll
        for (int cc = 0; cc < 3; ++cc)
            atomicAdd(&acc[dn * 9 + hh * 3 + cc], p * h[sn * 9 + hh * 3 + cc]);
    }
}

// =====================================================================
// Kernel 6: layer-1 epilogue — normalize, +bias, ELU.
// =====================================================================
__global__ void finish_elu(const float* __restrict__ acc, const float* __restrict__ s,
                           const float* __restrict__ b, float* __restrict__ t, int N) {
    int n = blockIdx.x * blockDim.x + threadIdx.x;
    if (n >= N) return;
    #pragma unroll
    for (int hh = 0; hh < HEADS; ++hh) {
        float inv = 1.0f / (s[n * 3 + hh] + GAT_EPS);
        #pragma unroll
        for (int cc = 0; cc < 3; ++cc) {
            float v = acc[n * 9 + hh * 3 + cc] * inv + b[hh * 3 + cc];
            t[n * 9 + hh * 3 + cc] = (v > 0.f) ? v : (expf(v) - 1.0f);
        }
    }
}

// =====================================================================
// Kernel 7: tiny layer-2 GEMM (9x9) — pure VALU, per node.
// =====================================================================
__global__ void gemm2(const float* __restrict__ t, const float* __restrict__ W2,
                      float* __restrict__ h2, int N) {
    __shared__ float Ws[81];
    if (threadIdx.x < 81) Ws[threadIdx.x] = W2[threadIdx.x];
    __syncthreads();
    int n = blockIdx.x * blockDim.x + threadIdx.x;
    if (n >= N) return;
    float tin[9];
    #pragma unroll
    for (int i = 0; i < 9; ++i) tin[i] = t[n * 9 + i];
    #pragma unroll
    for (int j = 0; j < 9; ++j) {
        float a = 0.f;
        #pragma unroll
        for (int i = 0; i < 9; ++i) a += tin[i] * Ws[i * 9 + j];
        h2[n * 9 + j] = a;
    }
}

// =====================================================================
// Kernel 8: layer-2 epilogue — normalize, head-mean, +bias, log_softmax.
// =====================================================================
__global__ void finish_out(const float* __restrict__ acc, const float* __restrict__ s,
                           const float* __restrict__ b2, float* __restrict__ out, int N) {
    int n = blockIdx.x * blockDim.x + threadIdx.x;
    if (n >= N) return;
    float v[NCLS];
    #pragma unroll
    for (int cc = 0; cc < NCLS; ++cc) {
        float acv = 0.f;
        #pragma unroll
        for (int hh = 0; hh < HEADS; ++hh)
            acv += acc[n * 9 + hh * 3 + cc] / (s[n * 3 + hh] + GAT_EPS);
        v[cc] = acv * (1.0f / HEADS) + b2[cc];
    }
    float mx = fmaxf(v[0], fmaxf(v[1], v[2]));
    float lse = logf(expf(v[0] - mx) + expf(v[1] - mx) + expf(v[2] - mx));
    #pragma unroll
    for (int cc = 0; cc < NCLS; ++cc)
        out[n * NCLS + cc] = v[cc] - mx - lse;
}

// =====================================================================
extern "C" void kernel_launch(void* const* d_in, const int* in_sizes, int n_in,
                              void* d_out, int out_size, void* d_ws, size_t ws_size,
                              hipStream_t stream) {
    const float* x        = (const float*)d_in[0];
    const int*   eidx     = (const int*)d_in[1];
    const float* W1       = (const float*)d_in[2];
    const float* att_src1 = (const float*)d_in[3];
    const float* att_dst1 = (const float*)d_in[4];
    const float* b1       = (const float*)d_in[5];
    const float* W2       = (const float*)d_in[6];
    const float* att_src2 = (const float*)d_in[7];
    const float* att_dst2 = (const float*)d_in[8];
    const float* b2       = (const float*)d_in[9];

    const int N = in_sizes[0] / IN_DIM;
    const int E = in_sizes[1] / 2;
    const int* src = eidx;
    const int* dst = eidx + E;

    // Workspace layout (layer 2 reuses layer-1 buffers): 39 N floats ~ 15.6 MB
    float*    h   = (float*)d_ws;               // [N,9]
    float*    as_ = h   + (size_t)N * 9;        // [N,3]
    float*    ad_ = as_ + (size_t)N * 3;        // [N,3]
    unsigned* m   = (unsigned*)(ad_ + (size_t)N * 3); // [N,3]
    float*    s   = (float*)(m + (size_t)N * 3);// [N,3]
    float*    acc = s   + (size_t)N * 3;        // [N,9]
    float*    t   = acc + (size_t)N * 9;        // [N,9]

    const int nb = (N + 255) / 256;
    const int eb = (E + 255) / 256;
    const int tiles = (N + 15) / 16;
    const int gb = (tiles + 7) / 8;             // 8 waves (tiles) per 256-thread block

    // ----- Layer 1 -----
    gemm1_wmma <<<gb, 256, 0, stream>>>(x, W1, h, N);
    node_att   <<<nb, 256, 0, stream>>>(h, att_src1, att_dst1, as_, ad_, m, N);
    edge_max   <<<eb, 256, 0, stream>>>(src, dst, as_, ad_, m, E);
    node_init  <<<nb, 256, 0, stream>>>(h, as_, ad_, m, s, acc, N);
    edge_scatter<<<eb, 256, 0, stream>>>(src, dst, h, as_, ad_, m, s, acc, E);
    finish_elu <<<nb, 256, 0, stream>>>(acc, s, b1, t, N);

    // ----- Layer 2 (reuses h/as_/ad_/m/s/acc) -----
    gemm2      <<<nb, 256, 0, stream>>>(t, W2, h, N);
    node_att   <<<nb, 256, 0, stream>>>(h, att_src2, att_dst2, as_, ad_, m, N);
    edge_max   <<<eb, 256, 0, stream>>>(src, dst, as_, ad_, m, E);
    node_init  <<<nb, 256, 0, stream>>>(h, as_, ad_, m, s, acc, N);
    edge_scatter<<<eb, 256, 0, stream>>>(src, dst, h, as_, ad_, m, s, acc, E);
    finish_out <<<nb, 256, 0, stream>>>(acc, s, b2, (float*)d_out, N);
}